// CrissCrossAttention_32813550141509
// MI455X (gfx1250) — compile-verified
//
#include <hip/hip_runtime.h>

#define BB  4
#define CC  64
#define CQKD 8
#define HH  256
#define WW  256
#define HWP (HH*WW)
#define NEG_BIG (-3.0e38f)

typedef __attribute__((ext_vector_type(2))) float v2f;
typedef __attribute__((ext_vector_type(8))) float v8f;

__device__ __forceinline__ v8f wmma_f32_16x16x4(v2f a, v2f b, v8f c) {
  // 8 args: (neg_a, A, neg_b, B, c_mod, C, reuse_a, reuse_b)
  return __builtin_amdgcn_wmma_f32_16x16x4_f32(false, a, false, b, (short)0, c, false, false);
}

// ---- CDNA5 async global<->LDS copies (ASYNCcnt path), 16B per lane-op ----
__device__ __forceinline__ unsigned lds_off_u32(const void* p) {
  return (unsigned)(uintptr_t)p;       // LDS aperture: low 32 bits = LDS byte address
}
__device__ __forceinline__ void async_cp16(void* ldst, const float* gsrc) {
  asm volatile("global_load_async_to_lds_b128 %0, %1, off"
               :: "v"(lds_off_u32(ldst)), "v"(gsrc) : "memory");
}
__device__ __forceinline__ void async_st16(float* gdst, const void* ldsrc) {
  asm volatile("global_store_async_from_lds_b128 %0, %1, off"
               :: "v"(gdst), "v"(lds_off_u32(ldsrc)) : "memory");
}
__device__ __forceinline__ void wait_async0() {
  asm volatile("s_wait_asynccnt 0x0" ::: "memory");
}

__device__ __forceinline__ float dot8(const float* __restrict__ a, const float* __restrict__ b) {
  float s = 0.f;
#pragma unroll
  for (int k = 0; k < CQKD; ++k) s = fmaf(a[k], b[k], s);
  return s;
}

// ---------------------------------------------------------------- K1: q/k/v projections
// q/k: VALU (K-out=8, too narrow for WMMA). v: (256px x 64) * (64 x 64) via f32 WMMA,
// D tiles staged in LDS and written back with async LDS->global b128 stores (coalesced).
__global__ void __launch_bounds__(256) k_proj(
    const float* __restrict__ x,  const float* __restrict__ Wq, const float* __restrict__ bq,
    const float* __restrict__ Wk, const float* __restrict__ bk,
    const float* __restrict__ Wv, const float* __restrict__ bv,
    float* __restrict__ qrw, float* __restrict__ krw, float* __restrict__ vT) {
  extern __shared__ float sm[];
  float* sWq = sm;                  // 512
  float* sWk = sWq + CQKD*CC;       // 512
  float* sWv = sWk + CQKD*CC;       // 4096   Wv[cv][c]
  float* Xs  = sWv + CC*CC;         // 16384  X[c][pix_local]
  float* Ds2 = Xs + CC*256;         // 2048   staging [32 px][64 c]
  const int t = threadIdx.x;
  const int pix0 = blockIdx.x * 256;
  const int b    = pix0 >> 16;
  const int rem0 = pix0 & (HWP - 1);

  for (int idx = t; idx < CQKD*CC; idx += 256) { sWq[idx] = Wq[idx]; sWk[idx] = Wk[idx]; }
  for (int idx = t; idx < CC*CC;  idx += 256)  sWv[idx] = Wv[idx];
  {
    const float* xb = x + (size_t)b * CC * HWP + rem0;
    for (int idx4 = t * 4; idx4 < CC * 256; idx4 += 1024) {
      int c = idx4 >> 8, p = idx4 & 255;
      async_cp16(&Xs[idx4], xb + (size_t)c * HWP + p);
    }
    wait_async0();
  }
  __syncthreads();

  // q/k (VALU)
#pragma unroll
  for (int kq = 0; kq < CQKD; ++kq) {
    float aq = bq[kq], ak = bk[kq];
#pragma unroll
    for (int c = 0; c < CC; ++c) {
      float xv = Xs[c*256 + t];
      aq = fmaf(sWq[kq*CC + c], xv, aq);
      ak = fmaf(sWk[kq*CC + c], xv, ak);
    }
    qrw[(size_t)(pix0 + t)*CQKD + kq] = aq;
    krw[(size_t)(pix0 + t)*CQKD + kq] = ak;
  }

  // v (WMMA): D[pix][cv] = sum_c X^T[pix][c] * Wv^T[c][cv] + bv[cv]
  const int lane = t & 31, wvid = t >> 5;
  const int am = lane & 15, ah = lane >> 4;
  for (int u = 0; u < 8; ++u) {
    // round u: 8 waves cover pix-tiles {2u, 2u+1} x all 4 c-tiles -> contiguous 8KB of vT
    const int pt = 2*u + (wvid >> 2);
    const int ct = wvid & 3;
    const float bias = bv[ct*16 + am];
    v8f acc;
#pragma unroll
    for (int r = 0; r < 8; ++r) acc[r] = bias;
#pragma unroll 4
    for (int ks = 0; ks < 16; ++ks) {
      const int ko = ks*4 + 2*ah;
      v2f a;  a.x  = Xs[ko*256 + pt*16 + am];       a.y  = Xs[(ko+1)*256 + pt*16 + am];
      v2f bb; bb.x = sWv[(ct*16 + am)*CC + ko];     bb.y = sWv[(ct*16 + am)*CC + ko + 1];
      acc = wmma_f32_16x16x4(a, bb, acc);
    }
    const int plocal = (wvid >> 2)*16 + 8*ah;       // pixel within 32-px round block
#pragma unroll
    for (int r = 0; r < 8; ++r)
      Ds2[(plocal + r)*CC + ct*16 + am] = acc[r];
    __syncthreads();
    {
      float* gbase = vT + (size_t)(pix0 + u*32)*CC + t*8;
      const float* lb = &Ds2[t*8];
      async_st16(gbase,     lb);
      async_st16(gbase + 4, lb + 4);
      wait_async0();                                // staging reused next round
    }
    __syncthreads();
  }
}

// ---------------------------------------------------------------- K2: column (eh) stats
__global__ void __launch_bounds__(256) k_colstats(
    const float* __restrict__ qrw, const float* __restrict__ krw,
    float* __restrict__ mb, float* __restrict__ sb) {
  __shared__ float Ks[HH*CQKD];
  const int b = blockIdx.x >> 8;
  const int w = blockIdx.x & 255;
  const int t = threadIdx.x;                     // pixel row i
  for (int idx4 = t*4; idx4 < HH*CQKD; idx4 += 1024) {
    int i2 = idx4 >> 3, kc = idx4 & 7;           // kc in {0,4}
    async_cp16(&Ks[idx4], krw + ((size_t)(b*HH + i2)*WW + w)*CQKD + kc);
  }
  wait_async0();
  __syncthreads();
  float qv[CQKD];
  const float* qp = qrw + ((size_t)(b*HH + t)*WW + w)*CQKD;
#pragma unroll
  for (int kc = 0; kc < CQKD; ++kc) qv[kc] = qp[kc];
  float m = NEG_BIG, s = 0.f;
  for (int j = 0; j < HH; ++j) {
    if (j == t) continue;                        // diagonal mask (-inf)
    float l = dot8(qv, &Ks[j*CQKD]);
    if (l > m) { s = s * __expf(m - l) + 1.f; m = l; }
    else       { s += __expf(l - m); }
  }
  mb[(size_t)(b*HH + t)*WW + w] = m;
  sb[(size_t)(b*HH + t)*WW + w] = s;
}

// ---------------------------------------------------------------- K3: row (ew) stats + exact LSE merge
__global__ void __launch_bounds__(256) k_rowstats(
    const float* __restrict__ qrw, const float* __restrict__ krw,
    float* __restrict__ mb, float* __restrict__ sb) {
  __shared__ float Ks[WW*CQKD];
  const int b = blockIdx.x >> 8;
  const int i = blockIdx.x & 255;
  const int t = threadIdx.x;                     // pixel col wq
  const size_t rbase = (size_t)(b*HH + i)*WW;
  for (int idx4 = t*4; idx4 < WW*CQKD; idx4 += 1024)
    async_cp16(&Ks[idx4], krw + rbase*CQKD + idx4);
  wait_async0();
  __syncthreads();
  float qv[CQKD];
#pragma unroll
  for (int kc = 0; kc < CQKD; ++kc) qv[kc] = qrw[(rbase + t)*CQKD + kc];
  float m2 = NEG_BIG, s2 = 0.f;
  for (int j = 0; j < WW; ++j) {
    float l = dot8(qv, &Ks[j*CQKD]);
    if (l > m2) { s2 = s2 * __expf(m2 - l) + 1.f; m2 = l; }
    else        { s2 += __expf(l - m2); }
  }
  float m1 = mb[rbase + t], s1 = sb[rbase + t];
  float M = fmaxf(m1, m2);
  float S = s1 * __expf(m1 - M) + s2 * __expf(m2 - M);
  mb[rbase + t] = M;
  sb[rbase + t] = S;
}

// ---------------------------------------------------------------- K4: row output  out = x + gamma*out_w
__global__ void __launch_bounds__(256) k_rowout(
    const float* __restrict__ x,   const float* __restrict__ qrw, const float* __restrict__ krw,
    const float* __restrict__ vT,  const float* __restrict__ mb,  const float* __restrict__ sb,
    const float* __restrict__ gammap, float* __restrict__ out) {
  extern __shared__ float sm[];
  float* Qs = sm;                   // [256][8]
  float* Ks = Qs + WW*CQKD;         // [256][8]
  float* Vs = Ks + WW*CQKD;         // [256][64]  B-matrix (j x c)
  float* Ps = Vs + WW*CC;           // [16][256]  prob tile (A-matrix)
  float* Ds = Ps + 16*WW;           // [16][64]   D tile
  float* mAll   = Ds + 16*CC;       // [256]
  float* invAll = mAll + WW;        // [256]
  const int b = blockIdx.x >> 8;
  const int i = blockIdx.x & 255;
  const int t = threadIdx.x;
  const int lane = t & 31, wvid = t >> 5;
  const int am = lane & 15, ah = lane >> 4;
  const float gamma = gammap[0];
  const size_t rbase = (size_t)(b*HH + i)*WW;

  for (int idx4 = t*4; idx4 < WW*CQKD; idx4 += 1024) {
    async_cp16(&Qs[idx4], qrw + rbase*CQKD + idx4);
    async_cp16(&Ks[idx4], krw + rbase*CQKD + idx4);
  }
  for (int idx4 = t*4; idx4 < WW*CC; idx4 += 1024)
    async_cp16(&Vs[idx4], vT + rbase*CC + idx4);
  mAll[t]   = mb[rbase + t];
  invAll[t] = 1.f / sb[rbase + t];
  wait_async0();
  __syncthreads();

  const int nt = wvid & 3, kh = wvid >> 2;       // wave -> N-tile, K-half (out GEMM)
  for (int mt = 0; mt < 16; ++mt) {
    // ---- logits via WMMA: (16 x 8) * (8 x 256), wave handles 2 N-tiles
#pragma unroll
    for (int u = 0; u < 2; ++u) {
      const int nt2 = wvid*2 + u;
      v8f lg = {};
#pragma unroll
      for (int ko4 = 0; ko4 < CQKD; ko4 += 4) {
        const int ko = ko4 + 2*ah;
        v2f a;  a.x  = Qs[(mt*16 + am)*CQKD + ko];  a.y  = Qs[(mt*16 + am)*CQKD + ko + 1];
        v2f bb; bb.x = Ks[(nt2*16 + am)*CQKD + ko]; bb.y = Ks[(nt2*16 + am)*CQKD + ko + 1];
        lg = wmma_f32_16x16x4(a, bb, lg);
      }
#pragma unroll
      for (int r = 0; r < 8; ++r) {
        const int M = r + 8*ah;
        const int row = mt*16 + M;
        Ps[M*WW + nt2*16 + am] = __expf(lg[r] - mAll[row]) * invAll[row];
      }
    }
    __syncthreads();
    // ---- D(16x64) = P(16x256) x V(256x64): 4 N-tiles x 2 K-halves over 8 waves
    v8f acc = {};
    const int kbase = kh * 128;
#pragma unroll 4
    for (int ks = 0; ks < 32; ++ks) {
      const int kk = kbase + ks*4 + 2*ah;
      v2f a;  a.x  = Ps[am*WW + kk];            a.y  = Ps[am*WW + kk + 1];
      v2f bb; bb.x = Vs[kk*CC + nt*16 + am];    bb.y = Vs[(kk+1)*CC + nt*16 + am];
      acc = wmma_f32_16x16x4(a, bb, acc);
    }
    if (kh == 1) {
#pragma unroll
      for (int r = 0; r < 8; ++r) Ds[(r + 8*ah)*CC + nt*16 + am] = acc[r];
    }
    __syncthreads();
    if (kh == 0) {
#pragma unroll
      for (int r = 0; r < 8; ++r) Ds[(r + 8*ah)*CC + nt*16 + am] += acc[r];
    }
    __syncthreads();
    // ---- out[b][c][i][mt*16 + w0..w0+3] = x + gamma * D  (b128 store)
    const int c  = t >> 2;
    const int w0 = (t & 3) * 4;
    const size_t obase = ((size_t)(b*CC + c)*HH + i)*WW + mt*16 + w0;
    float4 xv4 = *(const float4*)&x[obase];
    float4 o;
    o.x = fmaf(gamma, Ds[(w0+0)*CC + c], xv4.x);
    o.y = fmaf(gamma, Ds[(w0+1)*CC + c], xv4.y);
    o.z = fmaf(gamma, Ds[(w0+2)*CC + c], xv4.z);
    o.w = fmaf(gamma, Ds[(w0+3)*CC + c], xv4.w);
    *(float4*)&out[obase] = o;
    __syncthreads();
  }
}

// ---------------------------------------------------------------- K5: column output  out += gamma*out_h
__global__ void __launch_bounds__(256) k_colout(
    const float* __restrict__ qrw, const float* __restrict__ krw,
    const float* __restrict__ vT,  const float* __restrict__ mb, const float* __restrict__ sb,
    const float* __restrict__ gammap, float* __restrict__ out) {
  extern __shared__ float sm[];
  float* Qs = sm;
  float* Ks = Qs + HH*CQKD;
  float* Vs = Ks + HH*CQKD;         // [256][64]
  float* Ps = Vs + HH*CC;           // [16][256]
  float* Ds = Ps + 16*HH;           // [16][64]
  float* mAll   = Ds + 16*CC;       // [256]
  float* invAll = mAll + HH;        // [256]
  const int b = blockIdx.x >> 8;
  const int w = blockIdx.x & 255;
  const int t = threadIdx.x;
  const int lane = t & 31, wvid = t >> 5;
  const int am = lane & 15, ah = lane >> 4;
  const float gamma = gammap[0];

  for (int idx4 = t*4; idx4 < HH*CQKD; idx4 += 1024) {
    const int i2 = idx4 >> 3, kc = idx4 & 7;     // kc in {0,4}
    const size_t g = ((size_t)(b*HH + i2)*WW + w)*CQKD + kc;
    async_cp16(&Qs[idx4], qrw + g);
    async_cp16(&Ks[idx4], krw + g);
  }
  for (int idx4 = t*4; idx4 < HH*CC; idx4 += 1024) {
    const int j = idx4 >> 6, c = idx4 & 63;      // c multiple of 4
    async_cp16(&Vs[idx4], vT + ((size_t)(b*HH + j)*WW + w)*CC + c);
  }
  mAll[t]   = mb[(size_t)(b*HH + t)*WW + w];
  invAll[t] = 1.f / sb[(size_t)(b*HH + t)*WW + w];
  wait_async0();
  __syncthreads();

  const int nt = wvid & 3, kh = wvid >> 2;
  for (int mt = 0; mt < 16; ++mt) {
#pragma unroll
    for (int u = 0; u < 2; ++u) {
      const int nt2 = wvid*2 + u;
      v8f lg = {};
#pragma unroll
      for (int ko4 = 0; ko4 < CQKD; ko4 += 4) {
        const int ko = ko4 + 2*ah;
        v2f a;  a.x  = Qs[(mt*16 + am)*CQKD + ko];  a.y  = Qs[(mt*16 + am)*CQKD + ko + 1];
        v2f bb; bb.x = Ks[(nt2*16 + am)*CQKD + ko]; bb.y = Ks[(nt2*16 + am)*CQKD + ko + 1];
        lg = wmma_f32_16x16x4(a, bb, lg);
      }
#pragma unroll
      for (int r = 0; r < 8; ++r) {
        const int M = r + 8*ah;
        const int row = mt*16 + M;               // pixel i
        const int col = nt2*16 + am;             // j
        float l = lg[r];
        if (col == row) l = NEG_BIG;             // diagonal mask
        Ps[M*HH + col] = __expf(l - mAll[row]) * invAll[row];
      }
    }
    __syncthreads();
    v8f acc = {};
    const int kbase = kh * 128;
#pragma unroll 4
    for (int ks = 0; ks < 32; ++ks) {
      const int kk = kbase + ks*4 + 2*ah;
      v2f a;  a.x  = Ps[am*HH + kk];            a.y  = Ps[am*HH + kk + 1];
      v2f bb; bb.x = Vs[kk*CC + nt*16 + am];    bb.y = Vs[(kk+1)*CC + nt*16 + am];
      acc = wmma_f32_16x16x4(a, bb, acc);
    }
    if (kh == 1) {
#pragma unroll
      for (int r = 0; r < 8; ++r) Ds[(r + 8*ah)*CC + nt*16 + am] = acc[r];
    }
    __syncthreads();
    if (kh == 0) {
#pragma unroll
      for (int r = 0; r < 8; ++r) Ds[(r + 8*ah)*CC + nt*16 + am] += acc[r];
    }
    __syncthreads();
    const int c  = t >> 2;
    const int r0 = (t & 3) * 4;
#pragma unroll
    for (int u = 0; u < 4; ++u) {
      const int r = r0 + u;
      const size_t oi = ((size_t)(b*CC + c)*HH + mt*16 + r)*WW + w;
      out[oi] += gamma * Ds[r*CC + c];
    }
    __syncthreads();
  }
}

// ---------------------------------------------------------------- launch
extern "C" void kernel_launch(void* const* d_in, const int* in_sizes, int n_in,
                              void* d_out, int out_size, void* d_ws, size_t ws_size,
                              hipStream_t stream) {
  (void)in_sizes; (void)n_in; (void)out_size; (void)ws_size;
  const float* x  = (const float*)d_in[0];
  const float* Wq = (const float*)d_in[1];
  const float* bq = (const float*)d_in[2];
  const float* Wk = (const float*)d_in[3];
  const float* bk = (const float*)d_in[4];
  const float* Wv = (const float*)d_in[5];
  const float* bv = (const float*)d_in[6];
  const float* gm = (const float*)d_in[7];
  float* out = (float*)d_out;

  float* qrw = (float*)d_ws;                         // [B][H][W][8]
  float* krw = qrw + (size_t)BB*HWP*CQKD;            // [B][H][W][8]
  float* vT  = krw + (size_t)BB*HWP*CQKD;            // [B][H][W][64] (NHWC)
  float* mb  = vT  + (size_t)BB*HWP*CC;              // [B][H][W]
  float* sb  = mb  + (size_t)BB*HWP;                 // [B][H][W]

  const dim3 blk(256);
  const size_t smem_proj = (size_t)(CQKD*CC*2 + CC*CC + CC*256 + 32*CC) * sizeof(float);       // 92 KB
  const size_t smem_out  = (size_t)(WW*CQKD*2 + WW*CC + 16*WW + 16*CC + 2*WW) * sizeof(float); // 102 KB

  k_proj    <<<dim3((BB*HWP)/256), blk, smem_proj, stream>>>(x, Wq, bq, Wk, bk, Wv, bv, qrw, krw, vT);
  k_colstats<<<dim3(BB*WW),        blk, 0,         stream>>>(qrw, krw, mb, sb);
  k_rowstats<<<dim3(BB*HH),        blk, 0,         stream>>>(qrw, krw, mb, sb);
  k_rowout  <<<dim3(BB*HH),        blk, smem_out,  stream>>>(x, qrw, krw, vT, mb, sb, gm, out);
  k_colout  <<<dim3(BB*WW),        blk, smem_out,  stream>>>(qrw, krw, vT, mb, sb, gm, out);
}